// SpeechSynthesizer_11570641895508
// MI455X (gfx1250) — compile-verified
//
#include <hip/hip_runtime.h>

// ---------------------------------------------------------------------------
// SpeechSynthesizer — live computation is ONLY the GRU recurrence + mel proj.
//   (prenet/attention outputs are discarded by the reference; d == h always)
// Strategy: single persistent workgroup (16 wave32 waves on one WGP),
//   h + gate accumulators in LDS, weights pre-packed into WMMA-fragment order,
//   fp32 WMMA (V_WMMA_F32_16X16X4_F32) to preserve reference precision.
// Round-2 fix: branchless tile scatter — unified gacc[32][1024] accumulator
//   (r|z|nx|nh) with affine per-tile base, removing the saveexec/cndmask
//   divergence seen in the round-1 epilogue.
// ---------------------------------------------------------------------------

typedef float v2f __attribute__((ext_vector_type(2)));
typedef float v4f __attribute__((ext_vector_type(4)));
typedef float v8f __attribute__((ext_vector_type(8)));

#define BQ    32            // batch
#define HH    256           // hidden
#define SSEQ  1024          // sequence length
#define MM    80            // mel bins
#define NT_G  96            // 1536 / 16 N-tiles for g = h @ [Wih | Whh]
#define NT_M  5             // 80 / 16 N-tiles for mel = h @ Wm
#define KB2   32            // 256 / 8  (pairs of K=4 blocks)
#define BLOCK 512           // 16 waves
#define TPW   12            // g-tiles per wave (192 / 16)

// LDS row strides (dwords). Even strides keep ds_load_b64 8B-aligned.
// GA_STR: 8*GA_STR % 64 == 32 -> the two lane-groups (rows +8) hit disjoint banks.
#define H_STR  258
#define GA_STR 1036

#define G_DWORDS (NT_G * KB2 * 32 * 4)   // 393216 dwords = 1.5 MB
#define M_DWORDS (NT_M * KB2 * 32 * 4)   // 20480 dwords  = 80 KB

// ---------------------------------------------------------------------------
// Pack [Wih | Whh] (256x1536) and Wm (256x80) into per-lane WMMA B-fragment
// order: dword layout ((ntile*KB2 + kb2)*32 + lane)*4 + d, where
//   kblock = kb2*2 + d/2 ; v = d%2 ; K = kblock*4 + (lane/16)*2 + v ;
//   N = ntile*16 + lane%16.
// (Assumed B 4x16 layout: mirrors documented A 16x4 layout, transposed.)
// ---------------------------------------------------------------------------
__global__ void pack_weights_kernel(const float* __restrict__ Wih,
                                    const float* __restrict__ Whh,
                                    const float* __restrict__ Wm,
                                    float* __restrict__ packG,
                                    float* __restrict__ packM)
{
    int i = blockIdx.x * blockDim.x + threadIdx.x;
    if (i < G_DWORDS) {
        int nt   = i >> 12;          // / 4096
        int rem  = i & 4095;
        int kb2  = rem >> 7;
        int rem2 = rem & 127;
        int lane = rem2 >> 2;
        int d    = rem2 & 3;
        int k    = (kb2 * 2 + (d >> 1)) * 4 + ((lane >> 4) << 1) + (d & 1);
        int n    = nt * 16 + (lane & 15);      // 0..1535
        packG[i] = (n < 768) ? Wih[k * 768 + n] : Whh[k * 768 + (n - 768)];
    } else if (i < G_DWORDS + M_DWORDS) {
        int j    = i - G_DWORDS;
        int nt   = j >> 12;
        int rem  = j & 4095;
        int kb2  = rem >> 7;
        int rem2 = rem & 127;
        int lane = rem2 >> 2;
        int d    = rem2 & 3;
        int k    = (kb2 * 2 + (d >> 1)) * 4 + ((lane >> 4) << 1) + (d & 1);
        int n    = nt * 16 + (lane & 15);      // 0..79
        packM[j] = Wm[k * MM + n];
    }
}

// ---------------------------------------------------------------------------
// Persistent GRU kernel: one workgroup, 16 waves, 1024 sequential steps.
// Unified gate accumulator gacc[b][0..1023]:
//   [0,256)   r  pre-act  = h@Wih[:, 0:256]  + h@Whh[:, 0:256]  + bih+bhh
//   [256,512) z  pre-act  = h@Wih[:,256:512] + h@Whh[:,256:512] + bih+bhh
//   [512,768) nx          = h@Wih[:,512:768] + bih[512:768]
//   [768,1024) nh         = h@Whh[:,512:768] + bhh[512:768]
// ---------------------------------------------------------------------------
__global__ __launch_bounds__(BLOCK, 1)
void gru_persistent_kernel(const float* __restrict__ bih,
                           const float* __restrict__ bhh,
                           const float* __restrict__ bm,
                           const float* __restrict__ packG,
                           const float* __restrict__ packM,
                           float* __restrict__ out)
{
    extern __shared__ float lds[];
    float* hbuf    = lds;                       // [BQ][H_STR]   hidden state
    float* gacc    = hbuf + BQ * H_STR;         // [BQ][GA_STR]  unified gates
    float* biasAll = gacc + BQ * GA_STR;        // [1024]
    float* bmL     = biasAll + 4 * HH;          // [80]

    const int tid  = threadIdx.x;
    const int lane = tid & 31;
    const int wave = tid >> 5;
    const int lg   = lane >> 4;                 // lane group (K select)
    const int ln   = lane & 15;                 // M / N within tile

    // One-time setup: biases -> LDS, h = 0
    for (int i = tid; i < 4 * HH; i += BLOCK) {
        float v;
        if (i < 512)      v = bih[i] + bhh[i];      // r,z
        else if (i < 768) v = bih[i];               // nx (bih[512+j], j=i-512)
        else              v = bhh[i - 256];         // nh (bhh[512+j], j=i-768)
        biasAll[i] = v;
    }
    for (int i = tid; i < MM; i += BLOCK) bmL[i] = bm[i];
    for (int i = tid; i < BQ * H_STR; i += BLOCK) hbuf[i] = 0.0f;
    __syncthreads();

    for (int s = 0; s < SSEQ; ++s) {
        // ---- init gate accumulators with biases -------------------------
        for (int i = tid; i < BQ * 4 * HH; i += BLOCK) {
            int b = i >> 10, j = i & 1023;
            gacc[b * GA_STR + j] = biasAll[j];
        }
        __syncthreads();

        // ---- g = h @ [Wih | Whh] : 192 tiles of 16x16, 12 per wave ------
        for (int i = 0; i < TPW; ++i) {
            int t  = wave * TPW + i;
            int nt = t >> 1;                  // 0..95
            int mt = t & 1;                   // 0..1
            v8f acc = {};
            const v4f* bp = (const v4f*)packG + (size_t)(nt * KB2) * 32 + lane;
            const float* arow = hbuf + (mt * 16 + ln) * H_STR + lg * 2;
#pragma unroll 4
            for (int kb2 = 0; kb2 < KB2; ++kb2) {
                v4f bfrag = bp[kb2 * 32];                       // global_load_b128
                v2f a0 = *(const v2f*)(arow + kb2 * 8);         // ds_load_b64
                v2f a1 = *(const v2f*)(arow + kb2 * 8 + 4);     // ds_load_b64
                v2f b0 = { bfrag.x, bfrag.y };
                v2f b1 = { bfrag.z, bfrag.w };
                acc = __builtin_amdgcn_wmma_f32_16x16x4_f32(false, a0, false, b0,
                                                            (short)0, acc, false, false);
                acc = __builtin_amdgcn_wmma_f32_16x16x4_f32(false, a1, false, b1,
                                                            (short)0, acc, false, false);
            }
            // branchless scatter into unified accumulator:
            //   nt<48        : idx = nt*16 + ln           (Wih r/z/nx)
            //   48<=nt<80    : idx = nt*16 + ln - 768     (Whh r/z)
            //   nt>=80       : idx = nt*16 + ln - 512     (Whh n -> nh region)
            int sub  = (nt < 48) ? 0 : ((nt < 80) ? 768 : 512);
            int idx  = nt * 16 + ln - sub;
            float* dst = &gacc[(mt * 16 + lg * 8) * GA_STR + idx];
#pragma unroll
            for (int r = 0; r < 8; ++r)
                atomicAdd(&dst[r * GA_STR], acc[r]);            // ds_add_f32
        }
        __syncthreads();

        // ---- gates + hidden-state update --------------------------------
        for (int i = tid; i < BQ * HH; i += BLOCK) {
            int b = i >> 8, j = i & 255;
            const float* g = &gacc[b * GA_STR];
            float r = 1.0f / (1.0f + __expf(-g[j]));
            float z = 1.0f / (1.0f + __expf(-g[256 + j]));
            float n = tanhf(g[512 + j] + r * g[768 + j]);
            float hold = hbuf[b * H_STR + j];
            hbuf[b * H_STR + j] = (1.0f - z) * n + z * hold;
        }
        __syncthreads();

        // ---- mel = h_new @ Wm + bm : 10 tiles, waves 0..9 ---------------
        if (wave < 10) {
            int mt = wave & 1;
            int nt = wave >> 1;
            v8f acc = {};
            const v4f* bp = (const v4f*)packM + (size_t)(nt * KB2) * 32 + lane;
            const float* arow = hbuf + (mt * 16 + ln) * H_STR + lg * 2;
#pragma unroll 4
            for (int kb2 = 0; kb2 < KB2; ++kb2) {
                v4f bfrag = bp[kb2 * 32];
                v2f a0 = *(const v2f*)(arow + kb2 * 8);
                v2f a1 = *(const v2f*)(arow + kb2 * 8 + 4);
                v2f b0 = { bfrag.x, bfrag.y };
                v2f b1 = { bfrag.z, bfrag.w };
                acc = __builtin_amdgcn_wmma_f32_16x16x4_f32(false, a0, false, b0,
                                                            (short)0, acc, false, false);
                acc = __builtin_amdgcn_wmma_f32_16x16x4_f32(false, a1, false, b1,
                                                            (short)0, acc, false, false);
            }
            int m = nt * 16 + ln;
#pragma unroll
            for (int r = 0; r < 8; ++r) {
                int b = mt * 16 + r + lg * 8;
                out[(size_t)b * SSEQ * MM + (size_t)s * MM + m] = acc[r] + bmL[m];
            }
        }
        __syncthreads();
    }
}

// ---------------------------------------------------------------------------
extern "C" void kernel_launch(void* const* d_in, const int* in_sizes, int n_in,
                              void* d_out, int out_size, void* d_ws, size_t ws_size,
                              hipStream_t stream)
{
    // dict order: x, W1, b1, W2, b2, Wa, ba, Wih, bih, Whh, bhh, Wm, bm
    const float* Wih = (const float*)d_in[7];
    const float* bih = (const float*)d_in[8];
    const float* Whh = (const float*)d_in[9];
    const float* bhh = (const float*)d_in[10];
    const float* Wm  = (const float*)d_in[11];
    const float* bm  = (const float*)d_in[12];
    float* out = (float*)d_out;

    float* packG = (float*)d_ws;                    // 1.5 MB
    float* packM = packG + G_DWORDS;                // +80 KB  (ws_size >= ~1.6 MB)

    int tot = G_DWORDS + M_DWORDS;
    pack_weights_kernel<<<dim3((tot + 255) / 256), dim3(256), 0, stream>>>(
        Wih, Whh, Wm, packG, packM);

    size_t ldsBytes = (size_t)(BQ * H_STR + BQ * GA_STR + 4 * HH + MM)
                      * sizeof(float);              // ~170 KB
    (void)hipFuncSetAttribute((const void*)gru_persistent_kernel,
                              hipFuncAttributeMaxDynamicSharedMemorySize,
                              (int)ldsBytes);
    gru_persistent_kernel<<<dim3(1), dim3(BLOCK), ldsBytes, stream>>>(
        bih, bhh, bm, packG, packM, out);
}